// GraphConvolution_45140106281006
// MI455X (gfx1250) — compile-verified
//
#include <hip/hip_runtime.h>
#include <hip/hip_bf16.h>

typedef __attribute__((ext_vector_type(2))) float v2f;
typedef __attribute__((ext_vector_type(8))) float v8f;

#define F_DIM 256
#define SCAN_BLOCK 256
#define SCAN_ELEMS 1024   // 256 threads * 4 elements

// ---------------- column stats: partial sums ----------------
__global__ void gcn_stats_partial(const float* __restrict__ feat, int n,
                                  float* __restrict__ sum, float* __restrict__ sumsq,
                                  int rows_per_block) {
  int c = threadIdx.x;                       // one column per thread (256 threads)
  int r0 = blockIdx.x * rows_per_block;
  int r1 = min(n, r0 + rows_per_block);
  float s = 0.f, q = 0.f;
  for (int r = r0; r < r1; ++r) {            // coalesced: lane c reads row element c
    float v = feat[(size_t)r * F_DIM + c];
    s += v; q = fmaf(v, v, q);
  }
  atomicAdd(&sum[c], s);
  atomicAdd(&sumsq[c], q);
}

// ---------------- stats finalize: per-column affine a,b ----------------
__global__ void gcn_stats_finalize(const int* __restrict__ mode_p,
                                   const float* __restrict__ sum,
                                   const float* __restrict__ sumsq,
                                   int n, float* __restrict__ a, float* __restrict__ b) {
  int c = threadIdx.x;
  int mode = *mode_p;
  float mean = sum[c] / (float)n;
  float var  = (sumsq[c] - (float)n * mean * mean) / (float)(n - 1);  // ddof=1
  float sd   = sqrtf(fmaxf(var, 0.f)) + 1e-4f;
  float ac = 1.f, bc = 0.f;
  if (mode == 2)      { ac = 1.f / sd; bc = -mean * ac; }
  else if (mode == 1) { ac = 1.f;      bc = -mean; }
  a[c] = ac; b[c] = bc;
}

// ---------------- CSR build: degree histogram ----------------
__global__ void gcn_hist(const int* __restrict__ dst, int e, unsigned* __restrict__ deg) {
  int i = blockIdx.x * blockDim.x + threadIdx.x;
  if (i < e) atomicAdd(&deg[dst[i]], 1u);
}

// ---------------- CSR build: scan phase 1 (per-block exclusive scan) ----------------
__global__ void gcn_scan1(const unsigned* __restrict__ deg, int n,
                          unsigned* __restrict__ rowptr, unsigned* __restrict__ blocksum) {
  __shared__ unsigned tmp[SCAN_BLOCK];
  int t = threadIdx.x;
  int base = blockIdx.x * SCAN_ELEMS + t * 4;
  unsigned v0 = (base + 0 < n) ? deg[base + 0] : 0u;
  unsigned v1 = (base + 1 < n) ? deg[base + 1] : 0u;
  unsigned v2 = (base + 2 < n) ? deg[base + 2] : 0u;
  unsigned v3 = (base + 3 < n) ? deg[base + 3] : 0u;
  unsigned tsum = v0 + v1 + v2 + v3;
  tmp[t] = tsum;
  __syncthreads();
  for (int off = 1; off < SCAN_BLOCK; off <<= 1) {   // Hillis-Steele inclusive scan
    unsigned x = (t >= off) ? tmp[t - off] : 0u;
    __syncthreads();
    tmp[t] += x;
    __syncthreads();
  }
  unsigned excl = tmp[t] - tsum;
  if (t == SCAN_BLOCK - 1) blocksum[blockIdx.x] = tmp[t];
  unsigned run = excl;
  if (base + 0 < n) rowptr[base + 0] = run;  run += v0;
  if (base + 1 < n) rowptr[base + 1] = run;  run += v1;
  if (base + 2 < n) rowptr[base + 2] = run;  run += v2;
  if (base + 3 < n) rowptr[base + 3] = run;
}

// ---------------- CSR build: scan phase 2 (scan the block sums) ----------------
__global__ void gcn_scan2(unsigned* blocksum, int nb) {
  if (threadIdx.x == 0 && blockIdx.x == 0) {
    unsigned acc = 0;
    for (int i = 0; i < nb; ++i) { unsigned v = blocksum[i]; blocksum[i] = acc; acc += v; }
  }
}

// ---------------- CSR build: scan phase 3 (add offsets, init cursor) ----------------
__global__ void gcn_scan3(unsigned* __restrict__ rowptr, const unsigned* __restrict__ blockoff,
                          int n, unsigned e, unsigned* __restrict__ cursor) {
  int t = threadIdx.x;
  int base = blockIdx.x * SCAN_ELEMS + t * 4;
  unsigned off = blockoff[blockIdx.x];
#pragma unroll
  for (int i = 0; i < 4; ++i) {
    int idx = base + i;
    if (idx < n) { unsigned r = rowptr[idx] + off; rowptr[idx] = r; cursor[idx] = r; }
  }
  if (blockIdx.x == 0 && t == 0) rowptr[n] = e;
}

// ---------------- CSR build: fill buckets ----------------
__global__ void gcn_fill(const int* __restrict__ dst, const int* __restrict__ src,
                         const float* __restrict__ ew, int e,
                         unsigned* __restrict__ cursor,
                         int* __restrict__ csr_src, float* __restrict__ csr_w) {
  int i = blockIdx.x * blockDim.x + threadIdx.x;
  if (i < e) {
    unsigned pos = atomicAdd(&cursor[dst[i]], 1u);
    csr_src[pos] = src[i];
    csr_w[pos]   = ew[i];
  }
}

// ---------------- SpMM: one wave per destination node ----------------
__global__ void gcn_spmm(const float* __restrict__ feat,
                         const unsigned* __restrict__ rowptr,
                         const int* __restrict__ csr_src,
                         const float* __restrict__ csr_w,
                         const float* __restrict__ a,
                         const float* __restrict__ b,
                         int n, float* __restrict__ support) {
  int wave = threadIdx.x >> 5;
  int lane = threadIdx.x & 31;
  int node = blockIdx.x * (blockDim.x >> 5) + wave;
  if (node >= n) return;
  unsigned p0 = rowptr[node], p1 = rowptr[node + 1];
  int cbase = lane * 8;                       // lane owns 8 consecutive columns
  float4 acc0 = {0.f, 0.f, 0.f, 0.f};
  float4 acc1 = {0.f, 0.f, 0.f, 0.f};
  float wsum = 0.f;
  for (unsigned p = p0; p < p1; ++p) {
    int s = csr_src[p];
    float w = csr_w[p];
    if (p + 4 < p1) {                         // hide L2 gather latency
      int s2 = csr_src[p + 4];
      __builtin_prefetch(feat + (size_t)s2 * F_DIM + cbase, 0, 0);
    }
    const float4* fp = (const float4*)(feat + (size_t)s * F_DIM + cbase);
    float4 x0 = fp[0];
    float4 x1 = fp[1];
    acc0.x = fmaf(w, x0.x, acc0.x); acc0.y = fmaf(w, x0.y, acc0.y);
    acc0.z = fmaf(w, x0.z, acc0.z); acc0.w = fmaf(w, x0.w, acc0.w);
    acc1.x = fmaf(w, x1.x, acc1.x); acc1.y = fmaf(w, x1.y, acc1.y);
    acc1.z = fmaf(w, x1.z, acc1.z); acc1.w = fmaf(w, x1.w, acc1.w);
    wsum += w;
  }
  // support = a_c * (sum w*x) + b_c * (sum w)   (normalization folded in)
  float4 a0 = *(const float4*)(a + cbase);
  float4 a1 = *(const float4*)(a + cbase + 4);
  float4 b0 = *(const float4*)(b + cbase);
  float4 b1 = *(const float4*)(b + cbase + 4);
  float4 o0, o1;
  o0.x = fmaf(a0.x, acc0.x, b0.x * wsum); o0.y = fmaf(a0.y, acc0.y, b0.y * wsum);
  o0.z = fmaf(a0.z, acc0.z, b0.z * wsum); o0.w = fmaf(a0.w, acc0.w, b0.w * wsum);
  o1.x = fmaf(a1.x, acc1.x, b1.x * wsum); o1.y = fmaf(a1.y, acc1.y, b1.y * wsum);
  o1.z = fmaf(a1.z, acc1.z, b1.z * wsum); o1.w = fmaf(a1.w, acc1.w, b1.w * wsum);
  float4* op = (float4*)(support + (size_t)node * F_DIM + cbase);
  op[0] = o0;
  op[1] = o1;
}

// ---------------- GEMM: support @ W + bias via V_WMMA_F32_16X16X4_F32 ----------------
// block = 128 threads (4 waves). Each wave computes a 64x16 C strip (4 M-tiles,
// one N-tile), so the B fragment is reused across 4 WMMAs (1.5 loads/WMMA).
// grid.x covers M in 64-row strips, grid.y covers N in 64-col groups (wave -> +16).
__global__ void gcn_gemm_wmma(const float* __restrict__ A,   // [n, 256] support
                              const float* __restrict__ W,   // [256, 256]
                              const float* __restrict__ bias,
                              int n, float* __restrict__ out) {
  int wave = threadIdx.x >> 5;
  int lane = threadIdx.x & 31;
  int hv   = lane >> 4;          // 0: lanes 0-15 (K 0,1), 1: lanes 16-31 (K 2,3)
  int l16  = lane & 15;
  int m0 = blockIdx.x * 64;
  int n0 = blockIdx.y * 64 + wave * 16;
  int col = n0 + l16;

  // Clamp A-row pointers for the M tail; EXEC stays all-ones for every WMMA.
  int mr0 = m0 +      l16; if (mr0 > n - 1) mr0 = n - 1;
  int mr1 = m0 + 16 + l16; if (mr1 > n - 1) mr1 = n - 1;
  int mr2 = m0 + 32 + l16; if (mr2 > n - 1) mr2 = n - 1;
  int mr3 = m0 + 48 + l16; if (mr3 > n - 1) mr3 = n - 1;
  const float* arow0 = A + (size_t)mr0 * F_DIM;
  const float* arow1 = A + (size_t)mr1 * F_DIM;
  const float* arow2 = A + (size_t)mr2 * F_DIM;
  const float* arow3 = A + (size_t)mr3 * F_DIM;

  v8f c0 = {}, c1 = {}, c2 = {}, c3 = {};
#pragma unroll 2
  for (int k = 0; k < F_DIM; k += 4) {
    int ka = k + 2 * hv;
    // B 4x16 frag: lane holds (K=ka, N=col) and (K=ka+1, N=col) -- shared by 4 WMMAs
    v2f bv = { W[(size_t)ka * F_DIM + col], W[(size_t)(ka + 1) * F_DIM + col] };
    // A 16x4 frags: lane holds (M=l16, K=ka) and (M=l16, K=ka+1) -> one b64 load each
    v2f a0 = { arow0[ka], arow0[ka + 1] };
    v2f a1 = { arow1[ka], arow1[ka + 1] };
    v2f a2 = { arow2[ka], arow2[ka + 1] };
    v2f a3 = { arow3[ka], arow3[ka + 1] };
    c0 = __builtin_amdgcn_wmma_f32_16x16x4_f32(false, a0, false, bv, (short)0, c0, false, false);
    c1 = __builtin_amdgcn_wmma_f32_16x16x4_f32(false, a1, false, bv, (short)0, c1, false, false);
    c2 = __builtin_amdgcn_wmma_f32_16x16x4_f32(false, a2, false, bv, (short)0, c2, false, false);
    c3 = __builtin_amdgcn_wmma_f32_16x16x4_f32(false, a3, false, bv, (short)0, c3, false, false);
  }

  float bb = bias[col];
#pragma unroll
  for (int v = 0; v < 8; ++v) {      // C layout: VGPR v -> row v (lanes 0-15) / v+8
    int m = m0 + v + 8 * hv;
    if (m < n)      out[(size_t)m * F_DIM + col]        = c0[v] + bb;
    if (m + 16 < n) out[(size_t)(m + 16) * F_DIM + col] = c1[v] + bb;
    if (m + 32 < n) out[(size_t)(m + 32) * F_DIM + col] = c2[v] + bb;
    if (m + 48 < n) out[(size_t)(m + 48) * F_DIM + col] = c3[v] + bb;
  }
}

extern "C" void kernel_launch(void* const* d_in, const int* in_sizes, int n_in,
                              void* d_out, int out_size, void* d_ws, size_t ws_size,
                              hipStream_t stream) {
  const float* feat = (const float*)d_in[0];
  const float* ew   = (const float*)d_in[1];
  const float* W    = (const float*)d_in[2];
  const float* bias = (const float*)d_in[3];
  const int*   ei   = (const int*)d_in[4];
  const int*   mode = (const int*)d_in[5];

  int n = in_sizes[0] / F_DIM;     // 100000 nodes
  int e = in_sizes[1];             // 3200000 edges
  const int* dst = ei;             // edge_index row 0
  const int* src = ei + e;         // edge_index row 1

  // ---- workspace carve-up (256B aligned) ----
  char* base = (char*)d_ws;
  size_t off = 0;
  auto alloc = [&](size_t bytes) -> void* {
    void* p = base + off;
    off = (off + bytes + 255) & ~(size_t)255;
    return p;
  };
  float*    s_sum    = (float*)alloc(F_DIM * sizeof(float));
  float*    s_sumsq  = (float*)alloc(F_DIM * sizeof(float));
  float*    s_a      = (float*)alloc(F_DIM * sizeof(float));
  float*    s_b      = (float*)alloc(F_DIM * sizeof(float));
  unsigned* deg      = (unsigned*)alloc((size_t)n * 4);
  unsigned* rowptr   = (unsigned*)alloc(((size_t)n + 1) * 4);
  unsigned* cursor   = (unsigned*)alloc((size_t)n * 4);
  int nb = (n + SCAN_ELEMS - 1) / SCAN_ELEMS;
  unsigned* blocksum = (unsigned*)alloc((size_t)nb * 4);
  int*      csr_src  = (int*)alloc((size_t)e * 4);
  float*    csr_w    = (float*)alloc((size_t)e * 4);
  float*    support  = (float*)alloc((size_t)n * F_DIM * sizeof(float));
  (void)off; (void)ws_size;

  // zero only what is accumulated into (deterministic every call)
  hipMemsetAsync(s_sum,   0, F_DIM * sizeof(float), stream);
  hipMemsetAsync(s_sumsq, 0, F_DIM * sizeof(float), stream);
  hipMemsetAsync(deg,     0, (size_t)n * 4, stream);

  // 1) normalization stats
  const int rows_per_block = 256;
  int nstat = (n + rows_per_block - 1) / rows_per_block;
  gcn_stats_partial<<<nstat, 256, 0, stream>>>(feat, n, s_sum, s_sumsq, rows_per_block);
  gcn_stats_finalize<<<1, 256, 0, stream>>>(mode, s_sum, s_sumsq, n, s_a, s_b);

  // 2) CSR build
  gcn_hist<<<(e + 255) / 256, 256, 0, stream>>>(dst, e, deg);
  gcn_scan1<<<nb, SCAN_BLOCK, 0, stream>>>(deg, n, rowptr, blocksum);
  gcn_scan2<<<1, 32, 0, stream>>>(blocksum, nb);
  gcn_scan3<<<nb, SCAN_BLOCK, 0, stream>>>(rowptr, blocksum, n, (unsigned)e, cursor);
  gcn_fill<<<(e + 255) / 256, 256, 0, stream>>>(dst, src, ew, e, cursor, csr_src, csr_w);

  // 3) SpMM (one wave per node, L2-resident gathers)
  const int waves_per_block = 8;   // 256 threads
  gcn_spmm<<<(n + waves_per_block - 1) / waves_per_block, waves_per_block * 32, 0, stream>>>(
      feat, rowptr, csr_src, csr_w, s_a, s_b, n, support);

  // 4) WMMA GEMM: out = support @ W + bias  (64x16 C strip per wave)
  dim3 ggrid((n + 63) / 64, F_DIM / 64, 1);
  gcn_gemm_wmma<<<ggrid, 128, 0, stream>>>(support, W, bias, n, (float*)d_out);
}